// KnowledgeAwareCLIPLoss_42236708389749
// MI455X (gfx1250) — compile-verified
//
#include <hip/hip_runtime.h>
#include <math.h>

typedef float v2f __attribute__((ext_vector_type(2)));
typedef float v8f __attribute__((ext_vector_type(8)));

#define D_DIM 512
#define EPS_F 1e-8f
#define KW 0.1f

// -log_sigmoid(x) = log1p(exp(-|x|)) - min(x, 0)   (numerically stable)
__device__ __forceinline__ float neg_logsig(float x) {
    return log1pf(expf(-fabsf(x))) - fminf(x, 0.0f);
}

// Extract diag of 16x16 f32 C/D matrix onto lanes 0..15.
// Layout: D[m][n] in VGPR r, lane l with m = r + 8*(l>>4), n = l&15.
// Diag: i<8  -> VGPR i,   lane i
//       i>=8 -> VGPR i-8, lane i+16
__device__ __forceinline__ float diag_extract(v8f acc, int lane) {
    int e = (lane < 8) ? lane : (lane - 24);
    if (e < 0) e = 0;               // lanes 8..23: value unused
    float v = acc[0];
#pragma unroll
    for (int j = 1; j < 8; ++j) v = (e == j) ? acc[j] : v;
    int src = (lane < 8) ? lane : (lane + 16); // lanes 8..15 fetch from 24..31
    return __shfl(v, src & 31, 32);
}

__global__ __launch_bounds__(32)
void clip_tile_kernel(const float* __restrict__ img,
                      const float* __restrict__ txt,
                      const float* __restrict__ nam,
                      const float* __restrict__ gra,
                      const int*   __restrict__ ill,
                      float* __restrict__ partial, int M)
{
    const int lane = (int)threadIdx.x;
    const int tile = (int)blockIdx.x;
    const int row  = lane & 15;

    int p = tile * 16 + row;
    const float valid = (p < M) ? 1.0f : 0.0f;
    if (p >= M) p = (M > 0) ? (M - 1) : 0;

    const int e1 = ill[2 * p + 0];
    const int e2 = ill[2 * p + 1];
    const int koff = (lane >> 4) * 2;   // lanes 16..31 handle K%4 in {2,3}

    const float* pi1 = img + (size_t)e1 * D_DIM + koff;
    const float* pi2 = img + (size_t)e2 * D_DIM + koff;
    const float* pt1 = txt + (size_t)e1 * D_DIM + koff;
    const float* pt2 = txt + (size_t)e2 * D_DIM + koff;
    const float* pn1 = nam + (size_t)e1 * D_DIM + koff;
    const float* pn2 = nam + (size_t)e2 * D_DIM + koff;
    const float* pg1 = gra + (size_t)e1 * D_DIM + koff;
    const float* pg2 = gra + (size_t)e2 * D_DIM + koff;

    v8f acc_it = {};  // img1 . txt2
    v8f acc_ii = {};  // img1 . img2
    v8f acc_tt = {};  // txt1 . txt2
    v8f acc_nn = {};  // nam1 . nam2
    v8f acc_gg = {};  // gra1 . gra2

    float n_i1 = 0.f, n_i2 = 0.f, n_t1 = 0.f, n_t2 = 0.f;
    float n_n1 = 0.f, n_n2 = 0.f, n_g1 = 0.f, n_g2 = 0.f;

#pragma unroll 2
    for (int k = 0; k < D_DIM; k += 4) {
        v2f a_i1 = *(const v2f*)(pi1 + k);
        v2f a_i2 = *(const v2f*)(pi2 + k);
        v2f a_t1 = *(const v2f*)(pt1 + k);
        v2f a_t2 = *(const v2f*)(pt2 + k);
        v2f a_n1 = *(const v2f*)(pn1 + k);
        v2f a_n2 = *(const v2f*)(pn2 + k);
        v2f a_g1 = *(const v2f*)(pg1 + k);
        v2f a_g2 = *(const v2f*)(pg2 + k);

        // D = A(16x4 f32) x B(4x16 f32) + C ; diag accumulates the pair dots
        acc_it = __builtin_amdgcn_wmma_f32_16x16x4_f32(false, a_i1, false, a_t2,
                                                       (short)0, acc_it, false, false);
        acc_ii = __builtin_amdgcn_wmma_f32_16x16x4_f32(false, a_i1, false, a_i2,
                                                       (short)0, acc_ii, false, false);
        acc_tt = __builtin_amdgcn_wmma_f32_16x16x4_f32(false, a_t1, false, a_t2,
                                                       (short)0, acc_tt, false, false);
        acc_nn = __builtin_amdgcn_wmma_f32_16x16x4_f32(false, a_n1, false, a_n2,
                                                       (short)0, acc_nn, false, false);
        acc_gg = __builtin_amdgcn_wmma_f32_16x16x4_f32(false, a_g1, false, a_g2,
                                                       (short)0, acc_gg, false, false);

        // self-dots (norms^2) on the same registers, plain VALU fma
        n_i1 = fmaf(a_i1.x, a_i1.x, fmaf(a_i1.y, a_i1.y, n_i1));
        n_i2 = fmaf(a_i2.x, a_i2.x, fmaf(a_i2.y, a_i2.y, n_i2));
        n_t1 = fmaf(a_t1.x, a_t1.x, fmaf(a_t1.y, a_t1.y, n_t1));
        n_t2 = fmaf(a_t2.x, a_t2.x, fmaf(a_t2.y, a_t2.y, n_t2));
        n_n1 = fmaf(a_n1.x, a_n1.x, fmaf(a_n1.y, a_n1.y, n_n1));
        n_n2 = fmaf(a_n2.x, a_n2.x, fmaf(a_n2.y, a_n2.y, n_n2));
        n_g1 = fmaf(a_g1.x, a_g1.x, fmaf(a_g1.y, a_g1.y, n_g1));
        n_g2 = fmaf(a_g2.x, a_g2.x, fmaf(a_g2.y, a_g2.y, n_g2));
    }

    // lane r holds K%4 in {0,1}, lane r+16 holds {2,3}: combine halves
    n_i1 += __shfl_xor(n_i1, 16, 32);
    n_i2 += __shfl_xor(n_i2, 16, 32);
    n_t1 += __shfl_xor(n_t1, 16, 32);
    n_t2 += __shfl_xor(n_t2, 16, 32);
    n_n1 += __shfl_xor(n_n1, 16, 32);
    n_n2 += __shfl_xor(n_n2, 16, 32);
    n_g1 += __shfl_xor(n_g1, 16, 32);
    n_g2 += __shfl_xor(n_g2, 16, 32);

    const float d_it = diag_extract(acc_it, lane);
    const float d_ii = diag_extract(acc_ii, lane);
    const float d_tt = diag_extract(acc_tt, lane);
    const float d_nn = diag_extract(acc_nn, lane);
    const float d_gg = diag_extract(acc_gg, lane);

    const float ni1 = fmaxf(sqrtf(n_i1), EPS_F);
    const float ni2 = fmaxf(sqrtf(n_i2), EPS_F);
    const float nt1 = fmaxf(sqrtf(n_t1), EPS_F);
    const float nt2 = fmaxf(sqrtf(n_t2), EPS_F);
    const float nn1 = fmaxf(sqrtf(n_n1), EPS_F);
    const float nn2 = fmaxf(sqrtf(n_n2), EPS_F);
    const float ng1 = fmaxf(sqrtf(n_g1), EPS_F);
    const float ng2 = fmaxf(sqrtf(n_g2), EPS_F);

    const float c_it = d_it / (ni1 * nt2);
    const float c_ii = d_ii / (ni1 * ni2);
    const float c_tt = d_tt / (nt1 * nt2);
    const float c_nn = d_nn / (nn1 * nn2);
    const float c_gg = d_gg / (ng1 * ng2);

    float contrib = neg_logsig(c_it + c_ii + c_tt)
                  + KW * (neg_logsig(c_nn) + neg_logsig(c_gg));
    contrib *= valid;
    contrib = (lane < 16) ? contrib : 0.0f;  // lanes 16..31 carry no pairs

#pragma unroll
    for (int off = 16; off > 0; off >>= 1)
        contrib += __shfl_xor(contrib, off, 32);

    if (lane == 0) partial[tile] = contrib;
}

// Deterministic single-block tree reduction, then scale by 1/(3M).
__global__ __launch_bounds__(256)
void clip_reduce_kernel(const float* __restrict__ partial, int n,
                        float* __restrict__ out, float scale)
{
    __shared__ float sm[256];
    float s = 0.0f;
    for (int i = (int)threadIdx.x; i < n; i += 256) s += partial[i];
    sm[threadIdx.x] = s;
    __syncthreads();
#pragma unroll
    for (int off = 128; off > 0; off >>= 1) {
        if ((int)threadIdx.x < off) sm[threadIdx.x] += sm[threadIdx.x + off];
        __syncthreads();
    }
    if (threadIdx.x == 0) out[0] = sm[0] * scale;
}

extern "C" void kernel_launch(void* const* d_in, const int* in_sizes, int n_in,
                              void* d_out, int out_size, void* d_ws, size_t ws_size,
                              hipStream_t stream) {
    const float* img = (const float*)d_in[0];
    const float* txt = (const float*)d_in[1];
    const float* nam = (const float*)d_in[2];
    const float* gra = (const float*)d_in[3];
    const int*   ill = (const int*)d_in[4];

    const int M = in_sizes[4] / 2;
    const int nTiles = (M + 15) / 16;
    float* partial = (float*)d_ws;

    if (nTiles > 0) {
        clip_tile_kernel<<<nTiles, 32, 0, stream>>>(img, txt, nam, gra, ill, partial, M);
    }
    const float scale = (M > 0) ? (1.0f / (3.0f * (float)M)) : 0.0f;
    clip_reduce_kernel<<<1, 256, 0, stream>>>(partial, nTiles, (float*)d_out, scale);
}